// GNNModel_52639119179823
// MI455X (gfx1250) — compile-verified
//
#include <hip/hip_runtime.h>
#include <hip/hip_bf16.h>
#include <math.h>

typedef __attribute__((ext_vector_type(2))) float v2f;
typedef __attribute__((ext_vector_type(8))) float v8f;

#define DH 64  // hidden width, fixed by the model

// ---------------------------------------------------------------- utilities
__global__ void k_fill(float* __restrict__ p, float v, long n) {
  long i = (long)blockIdx.x * blockDim.x + threadIdx.x;
  if (i < n) p[i] = v;
}

__global__ void k_count_deg(const int* __restrict__ src, const int* __restrict__ dst,
                            float* __restrict__ deg, int E) {
  int e = blockIdx.x * blockDim.x + threadIdx.x;
  if (e < E) {
    atomicAdd(&deg[src[e]], 1.0f);
    atomicAdd(&deg[dst[e]], 1.0f);
  }
}

__global__ void k_rsqrt(float* __restrict__ d, int n) {
  int i = blockIdx.x * blockDim.x + threadIdx.x;
  if (i < n) d[i] = rsqrtf(d[i]);
}

// ------------------------------------------------- fp32 WMMA GEMM, N fixed = 64
// C[M x 64] = A[M x K] @ W[K x 64] (+ bias, optional ReLU)
// One wave computes a 16-row x 64-col stripe via 4 accumulators of 16x16.
// Uses V_WMMA_F32_16X16X4_F32 (native fp32 matrix op, matches reference precision).
//
// The weight is staged into LDS *pre-swizzled into WMMA B-fragment order*:
//   frag index i = (ks*4 + nt)*32 + lane  ->  v2f { W[kb][col], W[kb+1][col] }
//   where kb = 4*ks + (lane>>4)*2, col = nt*16 + (lane&15)
// so the inner loop is a single aligned ds_load_b64 per fragment (no register
// shuffling, lanes hit 64 consecutive banks).
template <int K>
__global__ void k_gemm_n64(const float* __restrict__ A, const float* __restrict__ W,
                           const float* __restrict__ bias, float* __restrict__ C,
                           int M, int do_relu) {
  __shared__ v2f Bs[K * 32];  // (K/4 k-steps) * 4 n-tiles * 32 lanes
  for (int i = threadIdx.x; i < K * 32; i += blockDim.x) {
    int ln = i & 31;
    int nt = (i >> 5) & 3;
    int ks = i >> 7;
    int kb = 4 * ks + (ln >> 4) * 2;
    int col = nt * 16 + (ln & 15);
    v2f t;
    t.x = W[(kb + 0) * DH + col];
    t.y = W[(kb + 1) * DH + col];
    Bs[i] = t;
  }
  __syncthreads();

  const int wave = threadIdx.x >> 5;
  const int lane = threadIdx.x & 31;
  const int row0 = (blockIdx.x * 4 + wave) * 16;
  if (row0 >= M) return;  // wave-uniform: EXEC stays all-ones for WMMA

  const int half = lane >> 4;  // 0: K sub-pair {k,k+1}; 1: {k+2,k+3}
  const int l    = lane & 15;

  // A fragment row for this lane (clamped for safety if M not multiple of 16)
  int ar = row0 + l;
  if (ar >= M) ar = M - 1;
  const float* arow = A + (size_t)ar * K + half * 2;

  v8f acc0 = {}, acc1 = {}, acc2 = {}, acc3 = {};

#pragma unroll
  for (int ks = 0; ks < K / 4; ++ks) {
    v2f a = *(const v2f*)(arow + ks * 4);
    const v2f* bf = &Bs[ks * 128 + lane];
    acc0 = __builtin_amdgcn_wmma_f32_16x16x4_f32(false, a, false, bf[0],
                                                 (short)0, acc0, false, false);
    acc1 = __builtin_amdgcn_wmma_f32_16x16x4_f32(false, a, false, bf[32],
                                                 (short)0, acc1, false, false);
    acc2 = __builtin_amdgcn_wmma_f32_16x16x4_f32(false, a, false, bf[64],
                                                 (short)0, acc2, false, false);
    acc3 = __builtin_amdgcn_wmma_f32_16x16x4_f32(false, a, false, bf[96],
                                                 (short)0, acc3, false, false);
  }

  // C/D layout: VGPR r -> row = row0 + r + half*8 ; lane l -> col = nt*16 + l
#pragma unroll
  for (int nt = 0; nt < 4; ++nt) {
    v8f c = (nt == 0) ? acc0 : (nt == 1) ? acc1 : (nt == 2) ? acc2 : acc3;
    const int col = nt * 16 + l;
    const float bv = bias ? bias[col] : 0.0f;
#pragma unroll
    for (int r = 0; r < 8; ++r) {
      int row = row0 + r + half * 8;
      if (row < M) {
        float v = c[r] + bv;
        if (do_relu) v = fmaxf(v, 0.0f);
        C[(size_t)row * DH + col] = v;
      }
    }
  }
}

// --------------------------------------------------------- graph aggregation
// agg[i] = xw[i] * dinv[i]^2   (self-loop term, also initializes agg)
__global__ void k_self(const float* __restrict__ xw, const float* __restrict__ dinv,
                       float* __restrict__ agg, int n) {
  long i = (long)blockIdx.x * blockDim.x + threadIdx.x;
  if (i < (long)n * DH) {
    int node = (int)(i >> 6);
    float di = dinv[node];
    agg[i] = xw[i] * di * di;
  }
}

// undirected scatter: for edge (s,d) add xw[s]*nm -> agg[d] and xw[d]*nm -> agg[s]
__global__ void k_scatter(const float* __restrict__ xw, const float* __restrict__ dinv,
                          const int* __restrict__ src, const int* __restrict__ dst,
                          float* __restrict__ agg, int E) {
  int e = blockIdx.x * 4 + (threadIdx.x >> 6);
  int f = threadIdx.x & 63;
  if (e >= E) return;
  int s = src[e], d = dst[e];
  float nm = dinv[s] * dinv[d];
  atomicAdd(&agg[(size_t)d * DH + f], xw[(size_t)s * DH + f] * nm);
  atomicAdd(&agg[(size_t)s * DH + f], xw[(size_t)d * DH + f] * nm);
}

__global__ void k_bias_relu(const float* __restrict__ agg, const float* __restrict__ b,
                            float* __restrict__ h, long total) {
  long i = (long)blockIdx.x * blockDim.x + threadIdx.x;
  if (i < total) h[i] = fmaxf(agg[i] + b[i & 63], 0.0f);
}

// ------------------------------------------------------------------ pooling
__global__ void k_pool(const float* __restrict__ h, const int* __restrict__ batch,
                       float* __restrict__ pooled, float* __restrict__ cnt, int n) {
  int node = blockIdx.x * 4 + (threadIdx.x >> 6);
  int f = threadIdx.x & 63;
  if (node >= n) return;
  int g = batch[node];
  atomicAdd(&pooled[(size_t)g * DH + f], h[(size_t)node * DH + f]);
  if (f == 0) atomicAdd(&cnt[g], 1.0f);
}

// ---------------------------------------------------------------- MLP head
// one block of 128 threads; thread g handles graph row g
__global__ void k_head(const float* __restrict__ pooled, const float* __restrict__ cnt,
                       const float* __restrict__ Wf1, const float* __restrict__ bf1,
                       const float* __restrict__ Wf2, const float* __restrict__ bf2,
                       float* __restrict__ out, int G) {
  __shared__ float sW1[DH * 32];
  __shared__ float sb1[32];
  __shared__ float sW2[32 * 10];
  __shared__ float sb2[10];
  for (int i = threadIdx.x; i < DH * 32; i += blockDim.x) sW1[i] = Wf1[i];
  for (int i = threadIdx.x; i < 32; i += blockDim.x) sb1[i] = bf1[i];
  for (int i = threadIdx.x; i < 320; i += blockDim.x) sW2[i] = Wf2[i];
  for (int i = threadIdx.x; i < 10; i += blockDim.x) sb2[i] = bf2[i];
  __syncthreads();

  int g = blockIdx.x * blockDim.x + threadIdx.x;
  if (g >= G) return;

  float inv = 1.0f / fmaxf(cnt[g], 1.0f);
  float m[DH];
#pragma unroll
  for (int k = 0; k < DH; ++k) m[k] = pooled[(size_t)g * DH + k] * inv;

  float hg[32];
#pragma unroll
  for (int j = 0; j < 32; ++j) {
    float s = sb1[j];
    for (int k = 0; k < DH; ++k) s += m[k] * sW1[k * 32 + j];
    hg[j] = fmaxf(s, 0.0f);
  }

  float lg[10];
#pragma unroll
  for (int c = 0; c < 10; ++c) {
    float s = sb2[c];
    for (int j = 0; j < 32; ++j) s += hg[j] * sW2[j * 10 + c];
    lg[c] = s;
  }

  float mx = lg[0];
#pragma unroll
  for (int c = 1; c < 10; ++c) mx = fmaxf(mx, lg[c]);
  float se = 0.0f;
#pragma unroll
  for (int c = 0; c < 10; ++c) se += expf(lg[c] - mx);
  float lse = mx + logf(se);
#pragma unroll
  for (int c = 0; c < 10; ++c) out[g * 10 + c] = lg[c] - lse;
}

// ------------------------------------------------------------------ launcher
extern "C" void kernel_launch(void* const* d_in, const int* in_sizes, int n_in,
                              void* d_out, int out_size, void* d_ws, size_t ws_size,
                              hipStream_t stream) {
  const float* x     = (const float*)d_in[0];
  const int*   ei    = (const int*)d_in[1];
  const int*   batch = (const int*)d_in[2];
  const float* W_in  = (const float*)d_in[3];
  const float* b_in  = (const float*)d_in[4];
  const float* W1    = (const float*)d_in[5];
  const float* bias1 = (const float*)d_in[6];
  const float* W2    = (const float*)d_in[7];
  const float* bias2 = (const float*)d_in[8];
  const float* Wf1   = (const float*)d_in[9];
  const float* bf1   = (const float*)d_in[10];
  const float* Wf2   = (const float*)d_in[11];
  const float* bf2   = (const float*)d_in[12];
  float* out = (float*)d_out;

  const int n = in_sizes[0] / 128;  // N_NODES
  const int E = in_sizes[1] / 2;    // directed edges (src<dst); undirected doubles
  const int G = out_size / 10;      // N_GRAPHS
  const int* src = ei;
  const int* dst = ei + E;

  // workspace layout (floats)
  float* ws = (float*)d_ws;
  size_t padn = ((size_t)n + 255) & ~(size_t)255;
  float* dinv   = ws;                          // n
  float* bufA   = dinv + padn;                 // n*64  (h)
  float* bufB   = bufA + (size_t)n * DH;       // n*64  (xw)
  float* bufC   = bufB + (size_t)n * DH;       // n*64  (agg)
  float* pooled = bufC + (size_t)n * DH;       // G*64
  float* cnt    = pooled + (size_t)G * DH;     // G

  const long tot = (long)n * DH;
  const int nb   = (n + 255) / 256;
  const int tb   = (int)((tot + 255) / 256);
  const int eb   = (E + 3) / 4;       // 4 edges per 256-thread block (64 feats each)
  const int pb   = (n + 3) / 4;
  const int gemmBlocks = (n + 63) / 64;  // 4 waves * 16 rows per block

  // degrees (incl. self loop) -> dinv = deg^{-1/2}
  k_fill<<<nb, 256, 0, stream>>>(dinv, 1.0f, n);
  k_count_deg<<<(E + 255) / 256, 256, 0, stream>>>(src, dst, dinv, E);
  k_rsqrt<<<nb, 256, 0, stream>>>(dinv, n);

  // h0 = relu(x @ W_in + b_in)  -> bufA
  k_gemm_n64<128><<<gemmBlocks, 128, 0, stream>>>(x, W_in, b_in, bufA, n, 1);

  // ---- GCN conv 1 ----
  k_gemm_n64<64><<<gemmBlocks, 128, 0, stream>>>(bufA, W1, nullptr, bufB, n, 0); // xw
  k_self<<<tb, 256, 0, stream>>>(bufB, dinv, bufC, n);                           // agg init
  k_scatter<<<eb, 256, 0, stream>>>(bufB, dinv, src, dst, bufC, E);              // neighbor agg
  k_bias_relu<<<tb, 256, 0, stream>>>(bufC, bias1, bufA, tot);                   // h1 -> bufA

  // ---- GCN conv 2 ----
  k_gemm_n64<64><<<gemmBlocks, 128, 0, stream>>>(bufA, W2, nullptr, bufB, n, 0);
  k_self<<<tb, 256, 0, stream>>>(bufB, dinv, bufC, n);
  k_scatter<<<eb, 256, 0, stream>>>(bufB, dinv, src, dst, bufC, E);
  k_bias_relu<<<tb, 256, 0, stream>>>(bufC, bias2, bufA, tot);                   // h2 -> bufA

  // ---- mean pool + head ----
  k_fill<<<(G * DH + 255) / 256, 256, 0, stream>>>(pooled, 0.0f, (long)G * DH);
  k_fill<<<1, 256, 0, stream>>>(cnt, 0.0f, G);
  k_pool<<<pb, 256, 0, stream>>>(bufA, batch, pooled, cnt, n);
  k_head<<<1, 128, 0, stream>>>(pooled, cnt, Wf1, bf1, Wf2, bf2, out, G);
}